// GAT_84559316124047
// MI455X (gfx1250) — compile-verified
//
#include <hip/hip_runtime.h>

#define DVEC 128

typedef float v2f __attribute__((ext_vector_type(2)));
typedef float v8f __attribute__((ext_vector_type(8)));

// ---------- ordered-int encoding for float atomicMax ----------
__device__ __forceinline__ int f32_ordered(float f) {
    int i = __float_as_int(f);
    return (i >= 0) ? i : (i ^ 0x7fffffff);
}
__device__ __forceinline__ float ordered_f32(int i) {
    return __int_as_float((i >= 0) ? i : (i ^ 0x7fffffff));
}

// ---------- K0: init denom/m, zero E_out, copy R -> output tail ----------
__global__ void gat_init(float* __restrict__ Eout, int nEout,
                         float* __restrict__ denom, int* __restrict__ m, int n,
                         const float* __restrict__ Rsrc, float* __restrict__ Rdst, int nR)
{
    int i = blockIdx.x * blockDim.x + threadIdx.x;
    if (i < nEout) Eout[i] = 0.0f;
    if (i < n) { denom[i] = 0.0f; m[i] = (int)0x807FFFFF; /* ordered(-inf) */ }
    if (i < nR) Rdst[i] = Rsrc[i];
}

// ---------- K1: WMMA projection  out_i = M @ Wi, out_j = M @ Wj ----------
// One wave handles 16 rows. A = 16x4 f32 tile of M; B col0 = Wi, col1 = Wj
// (staged in LDS). 32 x V_WMMA_F32_16X16X4_F32 accumulate 128-deep dots.
__global__ void gat_proj_wmma(const float* __restrict__ M, int nrows,
                              const float* __restrict__ Wi, const float* __restrict__ Wj,
                              float* __restrict__ out_i, float* __restrict__ out_j)
{
    __shared__ float sW[2 * DVEC];
    for (int j = threadIdx.x; j < 2 * DVEC; j += blockDim.x)
        sW[j] = (j < DVEC) ? Wi[j] : Wj[j - DVEC];
    __syncthreads();

    const int wave = threadIdx.x >> 5;
    const int lane = threadIdx.x & 31;
    const int rowBase = (blockIdx.x * (blockDim.x >> 5) + wave) * 16;
    if (rowBase >= nrows) return;              // wave-uniform: EXEC all-1s for WMMA

    const int half = lane >> 4;                // 0: K-pair {0,1}, 1: K-pair {2,3}
    const int l    = lane & 15;                // M (for A) / N (for B)
    const int rowA = min(rowBase + l, nrows - 1);   // clamp: branchless, stores guarded
    const float* __restrict__ rowPtr = M + (size_t)rowA * DVEC;

    const int   bbase = min(l, 1) * DVEC;      // lane 0 -> Wi, lane 1 -> Wj, others masked
    const float bmask = (l < 2) ? 1.0f : 0.0f;

    v8f c = {};
    #pragma unroll 8
    for (int k0 = 0; k0 < DVEC; k0 += 4) {
        const int ka = k0 + half * 2;
        v2f a;
        a.x = rowPtr[ka];
        a.y = rowPtr[ka + 1];
        v2f b;
        b.x = sW[bbase + ka]     * bmask;
        b.y = sW[bbase + ka + 1] * bmask;
        // D = A*B + C    (neg_a, A, neg_b, B, c_mod, C, reuse_a, reuse_b)
        c = __builtin_amdgcn_wmma_f32_16x16x4_f32(false, a, false, b, (short)0, c,
                                                  false, false);
    }

    // C/D layout: VGPR v, lanes 0-15 -> (M = v,     N = lane)
    //                     lanes16-31 -> (M = 8 + v, N = lane-16)
    if (l == 0 && out_i != nullptr) {
        #pragma unroll
        for (int v = 0; v < 8; ++v) {
            int row = rowBase + half * 8 + v;
            if (row < nrows) out_i[row] = c[v];
        }
    }
    if (l == 1 && out_j != nullptr) {
        #pragma unroll
        for (int v = 0; v < 8; ++v) {
            int row = rowBase + half * 8 + v;
            if (row < nrows) out_j[row] = c[v];
        }
    }
}

// ---------- K2: edge logits + segment max over h ----------
__global__ void gat_logits(const int* __restrict__ h, const int* __restrict__ r,
                           const int* __restrict__ t,
                           const float* __restrict__ ai, const float* __restrict__ aj,
                           const float* __restrict__ ar,
                           float* __restrict__ e, int* __restrict__ m, int nE)
{
    int i = blockIdx.x * blockDim.x + threadIdx.x;
    if (i >= nE) return;
    int hi = h[i];
    float v = ai[hi] + ar[r[i]] + aj[t[i]];
    v = (v >= 0.0f) ? v : 0.01f * v;           // leaky_relu
    e[i] = v;
    atomicMax(&m[hi], f32_ordered(v));
}

// ---------- K3: ex = exp(e - m[h]); denom[h] += ex ----------
__global__ void gat_expsum(const int* __restrict__ h, float* __restrict__ e,
                           const int* __restrict__ m, float* __restrict__ denom, int nE)
{
    int i = blockIdx.x * blockDim.x + threadIdx.x;
    if (i >= nE) return;
    int hi = h[i];
    float ex = expf(e[i] - ordered_f32(m[hi]));
    e[i] = ex;
    unsafeAtomicAdd(&denom[hi], ex);
}

// ---------- K4: scatter  Eout[h] += alpha * E[t]   (one wave per edge) ----------
__global__ void gat_scatter(const int* __restrict__ h, const int* __restrict__ t,
                            const float* __restrict__ ex, const float* __restrict__ denom,
                            const float* __restrict__ E, float* __restrict__ Eout, int nE)
{
    const int lane   = threadIdx.x & 31;
    const int wavesPerBlock = blockDim.x >> 5;
    const int wave   = blockIdx.x * wavesPerBlock + (threadIdx.x >> 5);
    const int nwaves = gridDim.x * wavesPerBlock;

    for (int i0 = wave; i0 < nE; i0 += nwaves) {
        // wave-uniform index -> scalar loads for edge metadata
        const int i  = __builtin_amdgcn_readfirstlane(i0);
        const int hi = h[i];
        const int ti = t[i];

        // prefetch next edge's gather row while we work on this one
        const int inext = i + nwaves;
        if (inext < nE) {
            const int tn = t[inext];
            __builtin_prefetch(E + (size_t)tn * DVEC + lane * 4, 0, 1);
        }

        const float alpha = ex[i] / denom[hi];
        const float4* src = (const float4*)(E + (size_t)ti * DVEC);
        float4 v = src[lane];                  // 32 lanes x float4 = 128 elems
        float* dst = Eout + (size_t)hi * DVEC + lane * 4;
        unsafeAtomicAdd(dst + 0, alpha * v.x);
        unsafeAtomicAdd(dst + 1, alpha * v.y);
        unsafeAtomicAdd(dst + 2, alpha * v.z);
        unsafeAtomicAdd(dst + 3, alpha * v.w);
    }
}

// ---------- K5: final ReLU ----------
__global__ void gat_relu(float* __restrict__ Eout, int nTot)
{
    int i = blockIdx.x * blockDim.x + threadIdx.x;
    if (i < nTot) Eout[i] = fmaxf(Eout[i], 0.0f);
}

extern "C" void kernel_launch(void* const* d_in, const int* in_sizes, int n_in,
                              void* d_out, int out_size, void* d_ws, size_t ws_size,
                              hipStream_t stream)
{
    const float* E  = (const float*)d_in[0];
    const float* R  = (const float*)d_in[1];
    const int*   h  = (const int*)  d_in[2];
    const int*   r  = (const int*)  d_in[3];
    const int*   t  = (const int*)  d_in[4];
    const float* Wi = (const float*)d_in[5];
    const float* Wj = (const float*)d_in[6];

    const int N  = in_sizes[0] / DVEC;   // 50000
    const int NR = in_sizes[1] / DVEC;   // 500
    const int NE = in_sizes[2];          // 800000

    float* Eout = (float*)d_out;                      // [N, D]
    float* Rout = Eout + (size_t)N * DVEC;            // [NR, D]

    // workspace layout (floats): ai | aj | ar | denom | e | m(int)
    float* ws    = (float*)d_ws;
    float* ai    = ws;
    float* aj    = ai + N;
    float* ar    = aj + N;
    float* denom = ar + NR;
    float* e     = denom + N;
    int*   m     = (int*)(e + NE);

    const int nEout = N * DVEC;
    const int nRout = NR * DVEC;

    // K0: init (largest range covers everything)
    {
        int blocks = (nEout + 255) / 256;
        gat_init<<<blocks, 256, 0, stream>>>(Eout, nEout, denom, m, N, R, Rout, nRout);
    }
    // K1: projections (WMMA). 256 threads = 8 waves = 128 rows / block.
    {
        int blocks = (N + 127) / 128;
        gat_proj_wmma<<<blocks, 256, 0, stream>>>(E, N, Wi, Wj, ai, aj);
        int blocksR = (NR + 127) / 128;
        gat_proj_wmma<<<blocksR, 256, 0, stream>>>(R, NR, Wi, Wj, nullptr, ar);
    }
    // K2: logits + segment max
    {
        int blocks = (NE + 255) / 256;
        gat_logits<<<blocks, 256, 0, stream>>>(h, r, t, ai, aj, ar, e, m, NE);
    }
    // K3: exp + segment sum
    {
        int blocks = (NE + 255) / 256;
        gat_expsum<<<blocks, 256, 0, stream>>>(h, e, m, denom, NE);
    }
    // K4: scatter-add messages (wave per edge)
    {
        int blocks = 4096;               // 32768 waves, ~25 edges each
        gat_scatter<<<blocks, 256, 0, stream>>>(h, t, e, denom, E, Eout, NE);
    }
    // K5: ReLU
    {
        int blocks = (nEout + 255) / 256;
        gat_relu<<<blocks, 256, 0, stream>>>(Eout, nEout);
    }
}